// MixtureOfExperts_23682449670302
// MI455X (gfx1250) — compile-verified
//
#include <hip/hip_runtime.h>

// MoE: N tokens, D=13 in/out, H=10 hidden, E=16 experts, K=2 top-k.
// - Gating (x@Wg, x@Wn) on the fp32 WMMA pipe: v_wmma_f32_16x16x4_f32,
//   16 tokens x 16 experts per wave-tile, fp32 to match reference top-k order.
// - x tiles staged via CDNA5 async copy (global_load_async_to_lds_b32),
//   double-buffered so the next tile's load overlaps this tile's compute.
// - Expert MLPs per-lane from LDS-resident weights; slot pair combined
//   with __shfl_xor.

typedef __attribute__((ext_vector_type(2))) float v2f;
typedef __attribute__((ext_vector_type(8))) float v8f;

#define MOE_D 13
#define MOE_H 10
#define MOE_E 16

// Issue one 16-token x tile (208 floats) as per-lane async copies into LDS.
// ldsBase = LDS byte offset of the destination (16x16-padded) tile.
__device__ __forceinline__ void prefetch_tile_async(const float* __restrict__ xsrc,
                                                    unsigned ldsBase, int lane) {
#pragma unroll
    for (int j = 0; j < 7; ++j) {
        int idx = lane + j * 32;
        if (idx < 16 * MOE_D) {                       // only j==6 diverges (lanes 0-15)
            int t = idx / MOE_D;
            int d = idx - t * MOE_D;
            unsigned loff = ldsBase + (unsigned)(t * 16 + d) * 4u;
            const float* g = xsrc + idx;              // 208 consecutive floats
            asm volatile("global_load_async_to_lds_b32 %0, %1, off"
                         :: "v"(loff), "v"(g) : "memory");
        }
    }
}

__global__ __launch_bounds__(256) void moe_wmma_kernel(
    const float* __restrict__ x,      // [N, 13]
    const float* __restrict__ noise,  // [N, 16]
    const float* __restrict__ Wg,     // [13, 16]
    const float* __restrict__ bg,     // [16]
    const float* __restrict__ Wn,     // [13, 16]
    const float* __restrict__ bn,     // [16]
    const float* __restrict__ W1,     // [16, 13, 10]
    const float* __restrict__ b1,     // [16, 10]
    const float* __restrict__ W2,     // [16, 10, 13]
    const float* __restrict__ b2,     // [16, 13]
    float* __restrict__ out,          // [N, 13]
    int nTok)
{
    constexpr int D = MOE_D, H = MOE_H, E = MOE_E;

    // ---- LDS: all expert weights + per-wave double-buffered tiles ----
    __shared__ float sW1[E * D * H];     // [e][d][h]
    __shared__ float sW2[E * H * D];     // [e][h][d]
    __shared__ float sB1[E * H];
    __shared__ float sB2[E * D];
    __shared__ float sX[8][2 * 256];     // per-wave x tiles: 2 buffers of 16x16 (pad 13->16)
    __shared__ float sNy[8][16 * 17];    // per-wave noisy logits, stride 17 (bank-safe)

    const int tid  = threadIdx.x;
    const int wave = tid >> 5;
    const int lane = tid & 31;
    const int half = lane >> 4;   // 0: lanes 0-15, 1: lanes 16-31
    const int n15  = lane & 15;

    for (int i = tid; i < E * D * H; i += 256) sW1[i] = W1[i];
    for (int i = tid; i < E * H * D; i += 256) sW2[i] = W2[i];
    for (int i = tid; i < E * H;     i += 256) sB1[i] = b1[i];
    for (int i = tid; i < E * D;     i += 256) sB2[i] = b2[i];

    // Zero the WMMA K-padding (cols 13..15) of both buffers ONCE; the async
    // copies only ever write cols 0..12, so the pads stay zero for all tiles.
    {
        float* xw = sX[wave];
        for (int i = lane; i < 32 * 3; i += 32) {   // 32 rows (2 bufs x 16) x 3 pad cols
            int r = i / 3, c = 13 + (i - r * 3);
            xw[r * 16 + c] = 0.f;
        }
    }
    __syncthreads();

    // ---- Build B fragments (constant across tiles): B is 4x16 (K x experts).
    // Layout: VGPR0 = rows K=2*half+0 ; VGPR1 = rows K=2*half+1 per 4-chunk.
    v2f Bg[4], Bn[4];
#pragma unroll
    for (int c = 0; c < 4; ++c) {
        int k0 = 4 * c + 2 * half;
        Bg[c].x = (k0     < D) ? Wg[k0 * E + n15]       : 0.f;
        Bg[c].y = (k0 + 1 < D) ? Wg[(k0 + 1) * E + n15] : 0.f;
        Bn[c].x = (k0     < D) ? Wn[k0 * E + n15]       : 0.f;
        Bn[c].y = (k0 + 1 < D) ? Wn[(k0 + 1) * E + n15] : 0.f;
    }
    const float bgE = bg[n15];
    const float bnE = bn[n15];

    float* nyT = sNy[wave];
    const unsigned sxBase = (unsigned)(size_t)(void*)&sX[wave][0];  // LDS byte offset

    const int numTiles = nTok >> 4;  // 16 tokens per tile
    const int stride   = gridDim.x * 8;
    int tile = blockIdx.x * 8 + wave;
    int buf  = 0;

    if (tile < numTiles)
        prefetch_tile_async(x + (long)tile * 16 * D, sxBase, lane);

    for (; tile < numTiles; tile += stride, buf ^= 1) {
        const int base = tile * 16;

        // Current buffer's async copies are complete (issued one compute ago).
        asm volatile("s_wait_asynccnt 0x0" ::: "memory");

        // Kick off next tile's copy into the other buffer; overlaps compute.
        const int nxt = tile + stride;
        if (nxt < numTiles)
            prefetch_tile_async(x + (long)nxt * 16 * D,
                                sxBase + (unsigned)((buf ^ 1) * 1024), lane);

        const float* xT = &sX[wave][buf * 256];

        // ---- A fragments: 16(tokens) x 4(K) fp32 per chunk ----
        v2f A[4];
#pragma unroll
        for (int c = 0; c < 4; ++c) {
            int k0 = 4 * c + 2 * half;
            A[c].x = xT[n15 * 16 + k0];
            A[c].y = xT[n15 * 16 + k0 + 1];
        }

        // ---- Gating matmuls on the WMMA pipe ----
        v8f Cg = {0.f, 0.f, 0.f, 0.f, 0.f, 0.f, 0.f, 0.f};
        v8f Cn = {0.f, 0.f, 0.f, 0.f, 0.f, 0.f, 0.f, 0.f};
#pragma unroll
        for (int c = 0; c < 4; ++c) {
            Cg = __builtin_amdgcn_wmma_f32_16x16x4_f32(false, A[c], false, Bg[c],
                                                       (short)0, Cg, false, false);
            Cn = __builtin_amdgcn_wmma_f32_16x16x4_f32(false, A[c], false, Bn[c],
                                                       (short)0, Cn, false, false);
        }

        // ---- Epilogue: noisy = logits + noise * softplus(noise_logits) ----
        // C layout: VGPR r <-> token (r + 8*half), lane&15 <-> expert.
#pragma unroll
        for (int r = 0; r < 8; ++r) {
            int tIn = r + 8 * half;
            float lg = Cg[r] + bgE;
            float nl = Cn[r] + bnE;
            float nz = noise[(long)(base + tIn) * E + n15];
            float sp = fmaxf(nl, 0.f) + log1pf(expf(-fabsf(nl)));  // stable softplus
            nyT[tIn * 17 + n15] = lg + nz * sp;
        }
        asm volatile("s_wait_dscnt 0x0" ::: "memory");

        // ---- Per-lane (token, slot): top-2, gates, expert MLP ----
        const int t16  = lane >> 1;   // token in tile
        const int slot = lane & 1;    // which of the top-2 experts

        float v1 = -3.402823466e38f, v2 = -3.402823466e38f;
        int i1 = 0, i2 = 0;
#pragma unroll
        for (int e = 0; e < E; ++e) {
            float v = nyT[t16 * 17 + e];
            if (v > v1)      { v2 = v1; i2 = i1; v1 = v; i1 = e; }
            else if (v > v2) { v2 = v;  i2 = e; }
        }
        const float ed = expf(v2 - v1);
        const float g1 = 1.f / (1.f + ed);
        const float g  = slot ? (ed * g1) : g1;
        const int   e  = slot ? i2 : i1;

        float xr[MOE_D];
#pragma unroll
        for (int d = 0; d < D; ++d) xr[d] = xT[t16 * 16 + d];

        const float* w1 = sW1 + e * (D * H);
        const float* w2 = sW2 + e * (H * D);
        float hb[MOE_H];
#pragma unroll
        for (int hh = 0; hh < H; ++hh) {
            float acc = sB1[e * H + hh];
#pragma unroll
            for (int d = 0; d < D; ++d) acc = fmaf(xr[d], w1[d * H + hh], acc);
            hb[hh] = fmaxf(acc, 0.f);
        }

        float* outp = out + (long)(base + t16) * D;
#pragma unroll
        for (int d = 0; d < D; ++d) {
            float acc = sB2[e * D + d];
#pragma unroll
            for (int hh = 0; hh < H; ++hh) acc = fmaf(hb[hh], w2[hh * D + d], acc);
            acc *= g;                        // gate-weight this slot
            acc += __shfl_xor(acc, 1, 32);   // combine the two slots of this token
            if (!slot) outp[d] = acc;
        }
    }
}

extern "C" void kernel_launch(void* const* d_in, const int* in_sizes, int n_in,
                              void* d_out, int out_size, void* d_ws, size_t ws_size,
                              hipStream_t stream) {
    const float* x     = (const float*)d_in[0];
    const float* noise = (const float*)d_in[1];
    const float* Wg    = (const float*)d_in[2];
    const float* bg    = (const float*)d_in[3];
    const float* Wn    = (const float*)d_in[4];
    const float* bn    = (const float*)d_in[5];
    const float* W1    = (const float*)d_in[6];
    const float* b1    = (const float*)d_in[7];
    const float* W2    = (const float*)d_in[8];
    const float* b2    = (const float*)d_in[9];
    float* out = (float*)d_out;

    const int nTok = in_sizes[0] / MOE_D;  // 2097152
    dim3 block(256);
    dim3 grid(1024);
    moe_wmma_kernel<<<grid, block, 0, stream>>>(x, noise, Wg, bg, Wn, bn,
                                                W1, b1, W2, b2, out, nTok);
}